// PeakExtractor_4844723110432
// MI455X (gfx1250) — compile-verified
//
#include <hip/hip_runtime.h>
#include <hip/hip_bf16.h>
#include <math.h>

// ---------------------------------------------------------------------------
// PeakExtractor for MI455X (gfx1250): streaming 5x5 NMS + top-3 via async
// global->LDS row pipeline (CDNA5 ASYNCcnt path), wave32 blocks.
// 4 rows per issue/wait/barrier iteration; horizontal maxima and center
// values kept in per-thread register rings (vertical pass is own-column).
// ---------------------------------------------------------------------------

#define HH 512
#define WW 512
#define NIMG 160          // 32*5
#define BANDS 4
#define BANDH 128         // HH / BANDS
#define RAWRING 16
#define NEG_INF (-__builtin_inff())

#if defined(__has_builtin)
#  if __has_builtin(__builtin_amdgcn_global_load_async_to_lds_b32)
#    define USE_ASYNC_LDS 1
#  endif
#  if __has_builtin(__builtin_amdgcn_s_wait_asynccnt)
#    define USE_WAIT_BUILTIN 1
#  endif
#endif

typedef __attribute__((address_space(1))) int gint_t;
typedef __attribute__((address_space(3))) int lint_t;

__device__ __forceinline__ void async_load_f32(const float* g, float* l) {
#if defined(USE_ASYNC_LDS) && defined(__AMDGCN__)
  __builtin_amdgcn_global_load_async_to_lds_b32((gint_t*)(g), (lint_t*)(l), 0, 0);
#else
  *l = *g;   // synchronous fallback (still correct with the waits/barriers)
#endif
}

template <int N>
__device__ __forceinline__ void wait_async() {
#if defined(USE_WAIT_BUILTIN) && defined(__AMDGCN__)
  __builtin_amdgcn_s_wait_asynccnt(N);
#elif defined(__AMDGCN__)
  asm volatile("s_wait_asynccnt %0" ::"i"(N) : "memory");
#endif
}

__device__ __forceinline__ bool better(float v, int i, float bv, int bi) {
  return (v > bv) || (v == bv && i < bi);
}

__device__ __forceinline__ void ins3(float v, int i,
                                     float& t0, int& i0,
                                     float& t1, int& i1,
                                     float& t2, int& i2) {
  if (better(v, i, t2, i2)) {
    t2 = v; i2 = i;
    if (better(t2, i2, t1, i1)) { float tv = t1; int ti = i1; t1 = t2; i1 = i2; t2 = tv; i2 = ti; }
    if (better(t1, i1, t0, i0)) { float tv = t0; int ti = i0; t0 = t1; i0 = i1; t1 = tv; i1 = ti; }
  }
}

// Process 4 output rows o = 4i..4i+3.  P = i&1 (compile-time) so all register
// ring indices fold to constants after unrolling (no scratch spills).
template <int P>
__device__ __forceinline__ void quad4(int i, int tid, int g0,
                                      float (&raw)[RAWRING][WW],
                                      float (&h)[8], float (&c)[8],
                                      float& t0, int& i0,
                                      float& t1, int& i1,
                                      float& t2, int& i2) {
  // Horizontal 5-max (clamped cols == -inf padding) for rows j = 4i+2..4i+5
#pragma unroll
  for (int k = 0; k < 4; ++k) {
    const int j = 4 * i + 2 + k;
    const float* row = &raw[(j + 2) & (RAWRING - 1)][0];
    const float ctr = row[tid];
    float m = ctr;
    m = fmaxf(m, row[tid - 2 < 0 ? 0 : tid - 2]);
    m = fmaxf(m, row[tid - 1 < 0 ? 0 : tid - 1]);
    m = fmaxf(m, row[tid + 1 > WW - 1 ? WW - 1 : tid + 1]);
    m = fmaxf(m, row[tid + 2 > WW - 1 ? WW - 1 : tid + 2]);
    h[(4 * P + 4 + k) & 7] = m;
    c[(4 * P + 4 + k) & 7] = ctr;
  }
  // Vertical 5-max (own column, registers) + NMS compare + top-3
#pragma unroll
  for (int r = 0; r < 4; ++r) {
    const int o = 4 * i + r;
    float vm = h[(4 * P + r + 0) & 7];
    vm = fmaxf(vm, h[(4 * P + r + 1) & 7]);
    vm = fmaxf(vm, h[(4 * P + r + 2) & 7]);
    vm = fmaxf(vm, h[(4 * P + r + 3) & 7]);
    vm = fmaxf(vm, h[(4 * P + r + 4) & 7]);
    const float ctr = c[(4 * P + r + 2) & 7];
    const float v = (ctr == vm) ? ctr : 0.0f;   // hm * (hm == hm_pool)
    ins3(v, (g0 + o) * WW + tid, t0, i0, t1, i1, t2, i2);
  }
}

// ---------------------------------------------------------------------------
// Stage 1: one block per (image, band). 512 threads; thread t owns column t.
// ---------------------------------------------------------------------------
__global__ __launch_bounds__(512, 1)
void nms_band_kernel(const float* __restrict__ hm,
                     float* __restrict__ wv, int* __restrict__ wi) {
  __shared__ float raw[RAWRING][WW];   // 32 KB row ring
  __shared__ float rv[512 * 3];        // 6 KB
  __shared__ int   ri[512 * 3];        // 6 KB

  const int blk  = blockIdx.x;
  const int img  = blk >> 2;
  const int band = blk & 3;
  const int g0   = band * BANDH;
  const int tid  = threadIdx.x;
  const float* base = hm + (size_t)img * (HH * WW);

  // Band-relative row j in [-2, BANDH+1]; clamped global row replicates the
  // -inf padding semantics for the vertical window max.
  auto issue = [&](int j) {
    int g = g0 + j;
    g = g < 0 ? 0 : (g > HH - 1 ? HH - 1 : g);
    async_load_f32(base + (size_t)g * WW + tid, &raw[(j + 2) & (RAWRING - 1)][tid]);
  };

  // Prologue: issue rows j = -2..9 (12 per-wave async instructions)
#pragma unroll
  for (int j = -2; j <= 9; ++j) issue(j);
  wait_async<8>();          // rows j=-2..1 (4 rows) guaranteed complete
  __syncthreads();

  float h[8], c[8];
#pragma unroll
  for (int k = 0; k < 8; ++k) { h[k] = NEG_INF; c[k] = 0.0f; }
#pragma unroll
  for (int j = -2; j <= 1; ++j) {      // register slots 0..3
    const float* row = &raw[(j + 2) & (RAWRING - 1)][0];
    const float ctr = row[tid];
    float m = ctr;
    m = fmaxf(m, row[tid - 2 < 0 ? 0 : tid - 2]);
    m = fmaxf(m, row[tid - 1 < 0 ? 0 : tid - 1]);
    m = fmaxf(m, row[tid + 1 > WW - 1 ? WW - 1 : tid + 1]);
    m = fmaxf(m, row[tid + 2 > WW - 1 ? WW - 1 : tid + 2]);
    h[j + 2] = m;
    c[j + 2] = ctr;
  }

  float t0 = NEG_INF, t1 = NEG_INF, t2 = NEG_INF;
  int   i0 = 0, i1 = 0, i2 = 0;

  // 32 iterations of 4 rows; unrolled x2 so the register-ring phase P = i&1
  // is a compile-time constant. Static ASYNCcnt schedule:
  //   iteration i (i<=29): issue rows 4i+10..4i+13, then wait asynccnt<=8
  //   => rows through j=4i+5 complete (exactly what quad4 consumes).
  for (int ii = 0; ii < BANDH / 8; ++ii) {
    {
      const int i = 2 * ii;
      if (i <= 29) {
#pragma unroll
        for (int k = 0; k < 4; ++k) issue(4 * i + 10 + k);
        wait_async<8>();
      } else {
        wait_async<0>();
      }
      __syncthreads();
      quad4<0>(i, tid, g0, raw, h, c, t0, i0, t1, i1, t2, i2);
    }
    {
      const int i = 2 * ii + 1;
      if (i <= 29) {
#pragma unroll
        for (int k = 0; k < 4; ++k) issue(4 * i + 10 + k);
        wait_async<8>();
      } else {
        wait_async<0>();
      }
      __syncthreads();
      quad4<1>(i, tid, g0, raw, h, c, t0, i0, t1, i1, t2, i2);
    }
  }

  // Block-wide top-3 tree reduction
  rv[tid * 3 + 0] = t0; rv[tid * 3 + 1] = t1; rv[tid * 3 + 2] = t2;
  ri[tid * 3 + 0] = i0; ri[tid * 3 + 1] = i1; ri[tid * 3 + 2] = i2;
  __syncthreads();
  for (int s = 256; s > 0; s >>= 1) {
    if (tid < s) {
      float a0 = rv[tid * 3], a1 = rv[tid * 3 + 1], a2 = rv[tid * 3 + 2];
      int   b0 = ri[tid * 3], b1 = ri[tid * 3 + 1], b2 = ri[tid * 3 + 2];
#pragma unroll
      for (int k = 0; k < 3; ++k)
        ins3(rv[(tid + s) * 3 + k], ri[(tid + s) * 3 + k], a0, b0, a1, b1, a2, b2);
      rv[tid * 3] = a0; rv[tid * 3 + 1] = a1; rv[tid * 3 + 2] = a2;
      ri[tid * 3] = b0; ri[tid * 3 + 1] = b1; ri[tid * 3 + 2] = b2;
    }
    __syncthreads();
  }
  if (tid < 3) {
    wv[blk * 3 + tid] = rv[tid];
    wi[blk * 3 + tid] = ri[tid];
  }
}

// ---------------------------------------------------------------------------
// Stage 2: merge the 4 band top-3 lists per image, softmax, emit outputs.
// Output layout (floats): coords[160*3*2] | scores[160*3] | indices[160*3]
// ---------------------------------------------------------------------------
__global__ void finalize_kernel(const float* __restrict__ wv,
                                const int* __restrict__ wi,
                                float* __restrict__ out) {
  const int img = blockIdx.x * blockDim.x + threadIdx.x;
  if (img >= NIMG) return;

  float t0 = NEG_INF, t1 = NEG_INF, t2 = NEG_INF;
  int   i0 = 0, i1 = 0, i2 = 0;
#pragma unroll
  for (int band = 0; band < BANDS; ++band) {
#pragma unroll
    for (int k = 0; k < 3; ++k) {
      const int s = (img * BANDS + band) * 3 + k;
      ins3(wv[s], wi[s], t0, i0, t1, i1, t2, i2);
    }
  }

  const float m  = t0;                 // sorted descending -> t0 is the max
  const float e0 = expf(t0 - m);
  const float e1 = expf(t1 - m);
  const float e2 = expf(t2 - m);
  const float inv = 1.0f / (e0 + e1 + e2);

  float* coords = out;                             // 960
  float* scores = out + NIMG * 3 * 2;              // 480
  float* inds   = out + NIMG * 3 * 2 + NIMG * 3;   // 480

  const int   idxs[3] = {i0, i1, i2};
  const float es[3]   = {e0, e1, e2};
#pragma unroll
  for (int k = 0; k < 3; ++k) {
    coords[img * 6 + 2 * k + 0] = (float)(idxs[k] >> 9);       // row = idx / 512
    coords[img * 6 + 2 * k + 1] = (float)(idxs[k] & (WW - 1)); // col = idx % 512
    scores[img * 3 + k] = es[k] * inv;
    inds[img * 3 + k]   = (float)idxs[k];
  }
}

// ---------------------------------------------------------------------------
extern "C" void kernel_launch(void* const* d_in, const int* in_sizes, int n_in,
                              void* d_out, int out_size, void* d_ws, size_t ws_size,
                              hipStream_t stream) {
  (void)in_sizes; (void)n_in; (void)out_size; (void)ws_size;
  const float* hm = (const float*)d_in[0];
  float* out = (float*)d_out;

  // workspace: per-band top-3 values then indices (640*3 each, ~15 KB)
  float* wv = (float*)d_ws;
  int*   wi = (int*)(wv + NIMG * BANDS * 3);

  nms_band_kernel<<<dim3(NIMG * BANDS), dim3(512), 0, stream>>>(hm, wv, wi);
  finalize_kernel<<<dim3((NIMG + 31) / 32), dim3(32), 0, stream>>>(wv, wi, out);
}